// LSTMForecaster_12996571037775
// MI455X (gfx1250) — compile-verified
//
#include <hip/hip_runtime.h>

// ---------------------------------------------------------------------------
// LSTM forecaster on gfx1250 (CDNA5) using fp32 WMMA (V_WMMA_F32_16X16X4_F32).
// gates[256,4096] = concat(x_t, h)[256,1088] @ W[1088,4096] + b, per step.
// One launch per timestep (512); h ping-pongs in workspace, c in place.
// K-contiguous LDS tiles -> ds_load_2addr_b64 fragments; register-staged LDS
// double buffering overlaps global loads with the WMMA stream. All staging
// state is in named scalars/vectors (no allocas -> no LDS promotion).
// ---------------------------------------------------------------------------

typedef __attribute__((ext_vector_type(2))) float v2f;
typedef __attribute__((ext_vector_type(8))) float v8f;

#define BATCH   256
#define SEQLEN  512
#define DIN     64
#define HID     1024
#define FANIN   (DIN + HID)     // 1088
#define HORIZON 24

#define BM 64                    // batch rows per block
#define BH 32                    // hidden cols per block (x4 gates = 128 gate cols)
#define KC 32                    // K-chunk
#define NCHUNK (FANIN / KC)      // 34
#define AS_STRIDE 36             // padded K-stride (bank-conflict-free b64 frags)
#define BS_STRIDE 36
#define WROW (4 * HID)           // 4096 floats per W row

__device__ __forceinline__ float sigmoid_f(float x) {
    return 1.0f / (1.0f + __expf(-x));
}
__device__ __forceinline__ float tanh_f(float x) {
    // 1 - 2/(e^{2x}+1): correct limits at both fp32 tails
    return 1.0f - 2.0f / (__expf(2.0f * x) + 1.0f);
}

__global__ void __launch_bounds__(256)
lstm_step_kernel(const float* __restrict__ x,
                 const float* __restrict__ W,
                 const float* __restrict__ bias,
                 const float* __restrict__ h_in,
                 float* __restrict__ h_out,
                 float* __restrict__ c_buf,
                 int t)
{
    __shared__ float As[2][BM * AS_STRIDE];    // [m][k], k contiguous
    __shared__ float Bs[2][128 * BS_STRIDE];   // [c][k], k contiguous (c = gate*32+col)

    const int tid    = threadIdx.x;
    const int lane   = tid & 31;
    const int wave   = tid >> 5;        // 0..7
    const int wave_m = wave & 3;        // 4 waves along batch
    const int wave_n = wave >> 2;       // 2 waves along hidden
    const int bm0    = blockIdx.x * BM;
    const int jH     = blockIdx.y * BH;

    const int half = lane >> 4;         // half-wave K phase (16x16x4 frag layout)
    const int l16  = lane & 15;
    const int colg = jH + wave_n * 16 + l16;   // global hidden column of C frag

    // --- A staging map: 2 x float4 per thread (64 rows x 32 k), named scalars
    const int am0 = tid >> 3;
    const int am1 = (tid + 256) >> 3;
    const int akv = (tid & 7) << 2;
    // --- B staging map: each thread owns one Bs column, half the k range
    const int bc  = tid & 127;          // 0..127: gate = bc>>5, col = bc&31
    const int bkh = (tid >> 7) << 4;    // 0 or 16
    const float* const wcol = W + (size_t)(bc >> 5) * HID + jH + (bc & 31);

    // staging registers (individually named: must stay in VGPRs)
    float4 pa0, pa1;
    float4 pb0, pb1, pb2, pb3;

    auto aload = [&](int k0, int m) -> float4 {
        const float* src;
        if (k0 < DIN) {     // DIN multiple of KC: no straddle
            src = x + (size_t)(bm0 + m) * (SEQLEN * DIN) + (size_t)t * DIN + (k0 + akv);
        } else {
            src = h_in + (size_t)(bm0 + m) * HID + (k0 - DIN + akv);
        }
        return *reinterpret_cast<const float4*>(src);
    };

    auto load_global = [&](int k0) {
        pa0 = aload(k0, am0);
        pa1 = aload(k0, am1);
        const float* src = wcol + (size_t)(k0 + bkh) * WROW;
        pb0.x = src[0 * (size_t)WROW];  pb0.y = src[1 * (size_t)WROW];
        pb0.z = src[2 * (size_t)WROW];  pb0.w = src[3 * (size_t)WROW];
        pb1.x = src[4 * (size_t)WROW];  pb1.y = src[5 * (size_t)WROW];
        pb1.z = src[6 * (size_t)WROW];  pb1.w = src[7 * (size_t)WROW];
        pb2.x = src[8 * (size_t)WROW];  pb2.y = src[9 * (size_t)WROW];
        pb2.z = src[10 * (size_t)WROW]; pb2.w = src[11 * (size_t)WROW];
        pb3.x = src[12 * (size_t)WROW]; pb3.y = src[13 * (size_t)WROW];
        pb3.z = src[14 * (size_t)WROW]; pb3.w = src[15 * (size_t)WROW];
    };

    auto store_lds = [&](int buf) {
        *reinterpret_cast<float4*>(&As[buf][am0 * AS_STRIDE + akv]) = pa0;
        *reinterpret_cast<float4*>(&As[buf][am1 * AS_STRIDE + akv]) = pa1;
        float* dst = &Bs[buf][bc * BS_STRIDE + bkh];
        *reinterpret_cast<float4*>(dst + 0)  = pb0;
        *reinterpret_cast<float4*>(dst + 4)  = pb1;
        *reinterpret_cast<float4*>(dst + 8)  = pb2;
        *reinterpret_cast<float4*>(dst + 12) = pb3;
    };

    // Accumulators i,f,g,o; bias folded into C init (uniform over rows).
    v8f acc[4];
    #pragma unroll
    for (int g = 0; g < 4; ++g) {
        const float bv = bias[g * HID + colg];
        #pragma unroll
        for (int r = 0; r < 8; ++r) acc[g][r] = bv;
    }

    const int arow = wave_m * 16 + l16;
    const int bcol = wave_n * 16 + l16;

    auto compute = [&](int buf) {
        const float* Ar = &As[buf][arow * AS_STRIDE];
        const float* Br = &Bs[buf][bcol * BS_STRIDE];
        #pragma unroll
        for (int kk = 0; kk < KC; kk += 4) {
            const int kp = kk + half * 2;
            const v2f a = *reinterpret_cast<const v2f*>(Ar + kp);   // ds_load_b64
            #pragma unroll
            for (int g = 0; g < 4; ++g) {
                const v2f bf =
                    *reinterpret_cast<const v2f*>(Br + g * 32 * BS_STRIDE + kp);
                acc[g] = __builtin_amdgcn_wmma_f32_16x16x4_f32(
                    false, a, false, bf, (short)0, acc[g], false, false);
            }
        }
    };

    // ---- software-pipelined main loop
    load_global(0);
    store_lds(0);
    __syncthreads();
    for (int ch = 0; ch < NCHUNK; ++ch) {
        const bool more = (ch + 1 < NCHUNK);
        if (more) load_global((ch + 1) * KC);       // in flight during compute
        compute(ch & 1);
        if (more) store_lds((ch + 1) & 1);
        __syncthreads();
    }

    // ---- pointwise LSTM update in-register (C frag: VGPR r -> M = r + 8*half)
    #pragma unroll
    for (int r = 0; r < 8; ++r) {
        const int rowg = bm0 + wave_m * 16 + r + half * 8;
        const size_t idx = (size_t)rowg * HID + colg;
        const float iv = sigmoid_f(acc[0][r]);
        const float fv = sigmoid_f(acc[1][r]);
        const float gv = tanh_f(acc[2][r]);
        const float ov = sigmoid_f(acc[3][r]);
        const float cn = fv * c_buf[idx] + iv * gv;
        c_buf[idx] = cn;
        h_out[idx] = ov * tanh_f(cn);
    }
}

__global__ void zero_kernel(float* __restrict__ p, int n) {
    const int i = blockIdx.x * blockDim.x + threadIdx.x;
    if (i < n) p[i] = 0.0f;
}

// out[256,24] = h_final @ Wout[1024,24] + bout
__global__ void __launch_bounds__(256)
proj_kernel(const float* __restrict__ h,
            const float* __restrict__ Wout,
            const float* __restrict__ bout,
            float* __restrict__ out)
{
    const int gid = blockIdx.x * blockDim.x + threadIdx.x;   // 0..6143
    const int b = gid / HORIZON;
    const int j = gid % HORIZON;
    if (b >= BATCH) return;
    float acc = bout[j];
    const float* hr = h + (size_t)b * HID;
    #pragma unroll 8
    for (int k = 0; k < HID; ++k) acc = fmaf(hr[k], Wout[k * HORIZON + j], acc);
    out[b * HORIZON + j] = acc;
}

extern "C" void kernel_launch(void* const* d_in, const int* in_sizes, int n_in,
                              void* d_out, int out_size, void* d_ws, size_t ws_size,
                              hipStream_t stream) {
    (void)in_sizes; (void)n_in; (void)out_size; (void)ws_size;
    const float* x    = (const float*)d_in[0];   // [256,512,64]
    const float* W    = (const float*)d_in[1];   // [1088,4096]
    const float* bias = (const float*)d_in[2];   // [4096]
    const float* Wout = (const float*)d_in[3];   // [1024,24]
    const float* bout = (const float*)d_in[4];   // [24]
    float* out = (float*)d_out;                  // [256,24]

    float* h0 = (float*)d_ws;                    // [256,1024]
    float* h1 = h0 + (size_t)BATCH * HID;        // [256,1024]
    float* c  = h1 + (size_t)BATCH * HID;        // [256,1024]

    {   // zero h0, h1, c (h0/c must be 0; h1 zeroed for determinism)
        const int n = 3 * BATCH * HID;
        zero_kernel<<<(n + 255) / 256, 256, 0, stream>>>(h0, n);
    }

    const dim3 grid(BATCH / BM, HID / BH);       // 4 x 32 = 128 blocks
    for (int t = 0; t < SEQLEN; ++t) {
        const float* hin  = (t & 1) ? h1 : h0;
        float*       hout = (t & 1) ? h0 : h1;
        lstm_step_kernel<<<grid, 256, 0, stream>>>(x, W, bias, hin, hout, c, t);
    }
    // SEQLEN even -> final h lands in h0.
    proj_kernel<<<(BATCH * HORIZON + 255) / 256, 256, 0, stream>>>(h0, Wout, bout, out);
}